// DQMixer_68917045232290
// MI455X (gfx1250) — compile-verified
//
#include <hip/hip_runtime.h>
#include <hip/hip_bf16.h>

// Shapes from the reference
#define BS 32
#define EP 32
#define NROWS (BS*EP)          // 1024
#define N_AGENTS 8
#define N_ATOM 51
#define EMBED 32
#define STATE_DIM 128
#define HYP 64
#define L1LEN (N_AGENTS*N_ATOM - N_AGENTS + 1)   // 401
#define L2LEN (EMBED*N_ATOM - EMBED + 1)         // 1601

typedef float v2f __attribute__((ext_vector_type(2)));
typedef float v8f __attribute__((ext_vector_type(8)));

// C51 projection scatter-add into a 51-bin LDS histogram (delta = 0.4)
__device__ __forceinline__ void proj_add(float* __restrict__ dst, float sup, float p) {
    float s = fminf(fmaxf(sup, -10.0f), 10.0f);
    float b = (s + 10.0f) * 2.5f;          // / 0.4
    float l = floorf(b);
    float u = ceilf(b);
    int li = min(max((int)l, 0), N_ATOM - 1);
    int ui = min(max((int)u, 0), N_ATOM - 1);
    float eq = (u == l) ? 1.0f : 0.0f;
    atomicAdd(&dst[li], p * (u - b + eq));
    atomicAdd(&dst[ui], p * (b - l));
}

// ---------------------------------------------------------------------------
// Kernel A: all MLP heads via V_WMMA_F32_16X16X4_F32.
// One workgroup = 16 rows; 8 waves; hidden activations staged through LDS.
// Stage 1 (K=128): H1=relu(S@hw1_w1^T+b) [64], H2=relu(S@hwf_w1^T+b) [64],
//                  B1=S@hb1_w^T+b [32] (to global), HV=relu(S@v_w1^T+b) [32]
// Stage 2 (K=64/32): w1=|H1@hw1_w2^T+b| [256], wf=|H2@hwf_w2^T+b| [32],
//                    v = HV@v_w2^T+b [1] (scalar dot, 1 col -> no WMMA)
// ---------------------------------------------------------------------------
__global__ __launch_bounds__(256) void dqmix_mlp_wmma(
    const float* __restrict__ S,
    const float* __restrict__ hw1_w1, const float* __restrict__ hw1_b1,
    const float* __restrict__ hw1_w2, const float* __restrict__ hw1_b2,
    const float* __restrict__ hb1_w,  const float* __restrict__ hb1_b,
    const float* __restrict__ hwf_w1, const float* __restrict__ hwf_b1,
    const float* __restrict__ hwf_w2, const float* __restrict__ hwf_b2,
    const float* __restrict__ v_w1,   const float* __restrict__ v_b1,
    const float* __restrict__ v_w2,   const float* __restrict__ v_b2,
    float* __restrict__ w1abs, float* __restrict__ b1lin,
    float* __restrict__ wfabs, float* __restrict__ vout)
{
    __shared__ float sS[16][132];   // 16 rows x 128 state (padded)
    __shared__ float sH1[16][68];
    __shared__ float sH2[16][68];
    __shared__ float sHV[16][36];

    const int tid  = threadIdx.x;
    const int wave = tid >> 5;
    const int lane = tid & 31;
    const int row0 = blockIdx.x * 16;

    for (int i = tid; i < 16 * STATE_DIM; i += 256)
        sS[i >> 7][i & 127] = S[(size_t)(row0 + (i >> 7)) * STATE_DIM + (i & 127)];
    __syncthreads();

    const int m16   = lane & 15;          // N index of B/C frag, M index of A frag
    const int khalf = (lane >> 4) << 1;   // 0 for lanes 0-15, 2 for lanes 16-31
    const int mbase = (lane >> 4) * 8;    // C frag row base

    // -------- stage 1: 12 output tiles of 16 cols, K = 128 --------
    for (int t = wave; t < 12; t += 8) {
        const float* W; const float* Bv; int n0, dst;
        if      (t < 4)  { W = hw1_w1; Bv = hw1_b1; n0 = t * 16;        dst = 0; }
        else if (t < 8)  { W = hwf_w1; Bv = hwf_b1; n0 = (t - 4) * 16;  dst = 1; }
        else if (t < 10) { W = hb1_w;  Bv = hb1_b;  n0 = (t - 8) * 16;  dst = 2; }
        else             { W = v_w1;   Bv = v_b1;   n0 = (t - 10) * 16; dst = 3; }

        v8f c = {};
        for (int k4 = 0; k4 < STATE_DIM; k4 += 4) {
            v2f a, b;
            a.x = sS[m16][k4 + khalf];
            a.y = sS[m16][k4 + khalf + 1];
            const float* wr = W + (size_t)(n0 + m16) * STATE_DIM + k4;
            b.x = wr[khalf];
            b.y = wr[khalf + 1];
            c = __builtin_amdgcn_wmma_f32_16x16x4_f32(false, a, false, b,
                                                      (short)0, c, false, false);
        }
        const float bias = Bv[n0 + m16];
        for (int v = 0; v < 8; ++v) {
            const float val = c[v] + bias;
            const int m = mbase + v;
            if      (dst == 0) sH1[m][n0 + m16] = fmaxf(val, 0.0f);
            else if (dst == 1) sH2[m][n0 + m16] = fmaxf(val, 0.0f);
            else if (dst == 2) b1lin[(size_t)(row0 + m) * EMBED + n0 + m16] = val;
            else               sHV[m][n0 + m16] = fmaxf(val, 0.0f);
        }
    }
    __syncthreads();

    // -------- stage 2: 16 tiles w1 (K=64, from H1) + 2 tiles wf (K=64, from H2) ----
    for (int t = wave; t < 18; t += 8) {
        const float* W; const float* Bv; int n0; int fromH1;
        if (t < 16) { W = hw1_w2; Bv = hw1_b2; n0 = t * 16;        fromH1 = 1; }
        else        { W = hwf_w2; Bv = hwf_b2; n0 = (t - 16) * 16; fromH1 = 0; }
        const float* Hbase = fromH1 ? &sH1[0][0] : &sH2[0][0];

        v8f c = {};
        for (int k4 = 0; k4 < HYP; k4 += 4) {
            v2f a, b;
            a.x = Hbase[m16 * 68 + k4 + khalf];
            a.y = Hbase[m16 * 68 + k4 + khalf + 1];
            const float* wr = W + (size_t)(n0 + m16) * HYP + k4;
            b.x = wr[khalf];
            b.y = wr[khalf + 1];
            c = __builtin_amdgcn_wmma_f32_16x16x4_f32(false, a, false, b,
                                                      (short)0, c, false, false);
        }
        const float bias = Bv[n0 + m16];
        for (int v = 0; v < 8; ++v) {
            const float val = fabsf(c[v] + bias);
            const int m = mbase + v;
            if (t < 16) w1abs[(size_t)(row0 + m) * 256   + n0 + m16] = val;
            else        wfabs[(size_t)(row0 + m) * EMBED + n0 + m16] = val;
        }
    }

    // -------- v head: 16 rows x (32-dot), single column --------
    if (wave == 0 && lane < 16) {
        float acc = v_b2[0];
        for (int k = 0; k < EMBED; ++k) acc += sHV[lane][k] * v_w2[k];
        vout[row0 + lane] = acc;
    }
}

// ---------------------------------------------------------------------------
// Kernel B: stage-1 distribution pipeline, one wave per (row, embed) task.
//  - project agent distributions through w1-scaled atoms (51 bins, LDS atomics)
//  - cascade full conv of the 8 agent distributions: 51 -> 401
//  - project through elu(support + b1) back to 51 atoms
//  - project through wf-scaled atoms -> xw[row][embed][51]
// Uniform loop trip counts across waves -> block-wide __syncthreads is legal.
// ---------------------------------------------------------------------------
__global__ __launch_bounds__(256) void dqmix_stage1(
    const float* __restrict__ aq,     // [1024, 8, 51]
    const float* __restrict__ w1abs,  // [1024, 256]
    const float* __restrict__ b1lin,  // [1024, 32]
    const float* __restrict__ wfabs,  // [1024, 32]
    float* __restrict__ xw)           // [1024, 32, 51]
{
    __shared__ float qA[8][408];
    __shared__ float qB[8][408];
    __shared__ float kb[8][64];
    __shared__ float xb[8][64];

    const int tid = threadIdx.x, wave = tid >> 5, lane = tid & 31;
    const int task = blockIdx.x * 8 + wave;
    const int r = task >> 5;
    const int e = task & 31;

    const float b1v = b1lin[(size_t)r * EMBED + e];
    const float wfv = wfabs[(size_t)r * EMBED + e];

    float* q  = qA[wave];
    float* qn = qB[wave];
    float* k0 = kb[wave];
    float* x0 = xb[wave];

    // agent 0 -> q
    for (int j = lane; j < 408; j += 32) q[j] = 0.0f;
    __syncthreads();
    {
        const float w = w1abs[(size_t)r * 256 + e * N_AGENTS + 0];
        for (int k = lane; k < N_ATOM; k += 32)
            proj_add(q, (-10.0f + 0.4f * k) * w, aq[((size_t)r * N_AGENTS + 0) * N_ATOM + k]);
    }
    __syncthreads();

    int L = N_ATOM;
    for (int a = 1; a < N_AGENTS; ++a) {
        for (int j = lane; j < 64; j += 32) k0[j] = 0.0f;
        __syncthreads();
        const float w = w1abs[(size_t)r * 256 + e * N_AGENTS + a];
        for (int k = lane; k < N_ATOM; k += 32)
            proj_add(k0, (-10.0f + 0.4f * k) * w, aq[((size_t)r * N_AGENTS + a) * N_ATOM + k]);
        __syncthreads();

        const int Lout = L + N_ATOM - 1;
        for (int n = lane; n < Lout; n += 32) {
            int mlo = n - (N_ATOM - 1); if (mlo < 0) mlo = 0;
            int mhi = n;                if (mhi > L - 1) mhi = L - 1;
            float acc = 0.0f;
            for (int m = mlo; m <= mhi; ++m) acc = fmaf(q[m], k0[n - m], acc);
            qn[n] = acc;
        }
        __syncthreads();
        float* t = q; q = qn; qn = t;
        L = Lout;
    }
    // L == 401 now in q

    for (int j = lane; j < 64; j += 32) x0[j] = 0.0f;
    __syncthreads();
    for (int j = lane; j < L1LEN; j += 32) {
        const float z = (-80.0f + 0.4f * j) + b1v;          // linspace(-80,80,401)+b1
        const float sup = (z > 0.0f) ? z : (__expf(z) - 1.0f);  // elu
        proj_add(x0, sup, q[j]);
    }
    __syncthreads();

    for (int j = lane; j < 64; j += 32) k0[j] = 0.0f;
    __syncthreads();
    for (int k = lane; k < N_ATOM; k += 32)
        proj_add(k0, (-10.0f + 0.4f * k) * wfv, x0[k]);
    __syncthreads();

    for (int k = lane; k < N_ATOM; k += 32)
        xw[((size_t)r * EMBED + e) * N_ATOM + k] = k0[k];
}

// ---------------------------------------------------------------------------
// Kernel C: stage-2 cascade over EMBED=32 distributions (51 -> 1601) and the
// final projection with support linspace(-320,320,1601)+v.  One block per row.
// ---------------------------------------------------------------------------
__global__ __launch_bounds__(256) void dqmix_stage2(
    const float* __restrict__ xw,    // [1024, 32, 51]
    const float* __restrict__ vout,  // [1024]
    float* __restrict__ out)         // [1024, 51]
{
    __shared__ float q0[1604];
    __shared__ float q1[1604];
    __shared__ float kb[64];
    __shared__ float ob[64];

    const int tid = threadIdx.x;
    const int r = blockIdx.x;

    float* q  = q0;
    float* qn = q1;

    for (int j = tid; j < N_ATOM; j += 256)
        q[j] = xw[(size_t)r * EMBED * N_ATOM + j];
    __syncthreads();

    int L = N_ATOM;
    for (int e = 1; e < EMBED; ++e) {
        for (int j = tid; j < N_ATOM; j += 256)
            kb[j] = xw[((size_t)r * EMBED + e) * N_ATOM + j];
        __syncthreads();

        const int Lout = L + N_ATOM - 1;
        for (int n = tid; n < Lout; n += 256) {
            int mlo = n - (N_ATOM - 1); if (mlo < 0) mlo = 0;
            int mhi = n;                if (mhi > L - 1) mhi = L - 1;
            float acc = 0.0f;
            for (int m = mlo; m <= mhi; ++m) acc = fmaf(q[m], kb[n - m], acc);
            qn[n] = acc;
        }
        __syncthreads();
        float* t = q; q = qn; qn = t;
        L = Lout;
    }
    // L == 1601 in q

    for (int j = tid; j < 64; j += 256) ob[j] = 0.0f;
    __syncthreads();
    const float vr = vout[r];
    for (int j = tid; j < L2LEN; j += 256)
        proj_add(ob, (-320.0f + 0.4f * j) + vr, q[j]);   // linspace(-320,320,1601)+v
    __syncthreads();

    for (int j = tid; j < N_ATOM; j += 256)
        out[(size_t)r * N_ATOM + j] = ob[j];
}

// ---------------------------------------------------------------------------
extern "C" void kernel_launch(void* const* d_in, const int* in_sizes, int n_in,
                              void* d_out, int out_size, void* d_ws, size_t ws_size,
                              hipStream_t stream) {
    (void)in_sizes; (void)n_in; (void)out_size; (void)ws_size;

    const float* aq     = (const float*)d_in[0];   // [32,32,8,51]
    const float* states = (const float*)d_in[1];   // [32,32,128]
    const float* hw1_w1 = (const float*)d_in[2];
    const float* hw1_b1 = (const float*)d_in[3];
    const float* hw1_w2 = (const float*)d_in[4];
    const float* hw1_b2 = (const float*)d_in[5];
    const float* hb1_w  = (const float*)d_in[6];
    const float* hb1_b  = (const float*)d_in[7];
    const float* hwf_w1 = (const float*)d_in[8];
    const float* hwf_b1 = (const float*)d_in[9];
    const float* hwf_w2 = (const float*)d_in[10];
    const float* hwf_b2 = (const float*)d_in[11];
    const float* v_w1   = (const float*)d_in[12];
    const float* v_b1   = (const float*)d_in[13];
    const float* v_w2   = (const float*)d_in[14];
    const float* v_b2   = (const float*)d_in[15];

    float* ws    = (float*)d_ws;
    float* w1abs = ws;                               // 1024*256
    float* b1lin = w1abs + (size_t)NROWS * 256;      // 1024*32
    float* wfabs = b1lin + (size_t)NROWS * EMBED;    // 1024*32
    float* vout  = wfabs + (size_t)NROWS * EMBED;    // 1024
    float* xw    = vout  + NROWS;                    // 1024*32*51

    float* out = (float*)d_out;                      // [32,32,51]

    dqmix_mlp_wmma<<<NROWS / 16, 256, 0, stream>>>(
        states,
        hw1_w1, hw1_b1, hw1_w2, hw1_b2,
        hb1_w, hb1_b,
        hwf_w1, hwf_b1, hwf_w2, hwf_b2,
        v_w1, v_b1, v_w2, v_b2,
        w1abs, b1lin, wfabs, vout);

    dqmix_stage1<<<(NROWS * EMBED) / 8, 256, 0, stream>>>(
        aq, w1abs, b1lin, wfabs, xw);

    dqmix_stage2<<<NROWS, 256, 0, stream>>>(
        xw, vout, out);
}